// GCNEncoder_59931973648610
// MI455X (gfx1250) — compile-verified
//
#include <hip/hip_runtime.h>
#include <math.h>

// GCN encoder for MI455X (gfx1250, wave32).
// d_in: x[f32 N*128], edge_index[int 2*E], W1[128*128], b1[128], W2[128*128], b2[128]
// d_out: concat(h2 [N*128], x [N*128])  (tuple return order)

#define D 128

typedef __attribute__((ext_vector_type(2))) float v2f;
typedef __attribute__((ext_vector_type(8))) float v8f;

// ---------------- degree / normalization ----------------

__global__ __launch_bounds__(256) void k_fill_one(float* deg, int n) {
  int i = blockIdx.x * 256 + threadIdx.x;
  if (i < n) deg[i] = 1.0f;  // self-loop contributes 1 to every node
}

__global__ __launch_bounds__(256) void k_deg_accum(const int* __restrict__ dst,
                                                   float* deg, int e) {
  int i = blockIdx.x * 256 + threadIdx.x;
  if (i < e) atomicAdd(&deg[dst[i]], 1.0f);
}

__global__ __launch_bounds__(256) void k_rsqrt(float* d, int n) {
  int i = blockIdx.x * 256 + threadIdx.x;
  if (i < n) d[i] = rsqrtf(d[i]);  // deg >= 1 always (self loops)
}

// ---------------- GEMM: H = X @ W  (fp32 WMMA 16x16x4) ----------------
// One wave per 16-row strip; 8 waves (256 thr) per block -> 128 rows/block.
// W (128x128 f32 = 64KB) staged in LDS once per block.

__global__ __launch_bounds__(256) void k_gemm128(const float* __restrict__ X,
                                                 const float* __restrict__ W,
                                                 float* __restrict__ H,
                                                 int nrows) {
  __shared__ float Ws[D * D];  // 64 KB of the 320 KB WGP LDS
  {
    const float4* Wv = reinterpret_cast<const float4*>(W);
    float4* Sv = reinterpret_cast<float4*>(Ws);
    for (int i = threadIdx.x; i < (D * D) / 4; i += 256) Sv[i] = Wv[i];
  }
  __syncthreads();

  const int wave = threadIdx.x >> 5;   // wave32
  const int lane = threadIdx.x & 31;
  const int half = lane >> 4;          // 0: lanes 0-15, 1: lanes 16-31
  const int l15  = lane & 15;

  const int strip = blockIdx.x * 8 + wave;
  const int row0  = strip * 16;

  // A-fragment row for this lane (16x4 layout: lanes 0-15 and 16-31 both M=0..15)
  int arow = row0 + l15;
  if (arow >= nrows) arow = nrows - 1;          // clamp: safe read, masked store
  const float* xrow = X + (size_t)arow * D;

  v8f acc[8] = {};  // 8 column tiles x 16x16 f32 accum = 64 VGPRs

  for (int k0 = 0; k0 < D; k0 += 4) {
    // A 16x4: v0 = K=k0 (lanes0-15) / K=k0+2 (lanes16-31); v1 = K+1 / K+3
    const int ak = k0 + half * 2;
    v2f a;
    a.x = xrow[ak];
    a.y = xrow[ak + 1];
#pragma unroll
    for (int j = 0; j < 8; ++j) {
      // B 4x16: v0 lanes0-15 = W[k0+0][col], lanes16-31 = W[k0+2][col]; v1 = rows +1
      const int col = j * 16 + l15;
      const int bk  = k0 + half * 2;
      v2f b;
      b.x = Ws[bk * D + col];
      b.y = Ws[(bk + 1) * D + col];
      acc[j] = __builtin_amdgcn_wmma_f32_16x16x4_f32(
          /*neg_a=*/false, a, /*neg_b=*/false, b,
          /*c_mod=*/(short)0, acc[j], /*reuse_a=*/false, /*reuse_b=*/false);
    }
  }

  // Epilogue. C/D layout: VGPR i -> row = row0 + half*8 + i, col = j*16 + lane%16.
  const int rbase = row0 + half * 8;
  if (row0 + 16 <= nrows) {
    // fast path: whole strip in range (all but the last partial strip)
    float* hb = H + (size_t)rbase * D + l15;
#pragma unroll
    for (int j = 0; j < 8; ++j) {
#pragma unroll
      for (int i = 0; i < 8; ++i) {
        hb[(size_t)i * D + j * 16] = acc[j][i];
      }
    }
  } else if (row0 < nrows) {
    // slow path: per-row guard (only the final partial strip takes this)
#pragma unroll
    for (int j = 0; j < 8; ++j) {
      const int col = j * 16 + l15;
#pragma unroll
      for (int i = 0; i < 8; ++i) {
        const int r = rbase + i;
        if (r < nrows) H[(size_t)r * D + col] = acc[j][i];
      }
    }
  }
}

// ---------------- aggregation ----------------

// out[i,c] = b[c] + H[i,c] * dis[i]^2   (bias + self-loop term)
__global__ __launch_bounds__(256) void k_agg_init(const float* __restrict__ H,
                                                  const float* __restrict__ dis,
                                                  const float* __restrict__ bias,
                                                  float* __restrict__ out,
                                                  int total) {
  int idx = blockIdx.x * 256 + threadIdx.x;
  if (idx >= total) return;
  int node = idx >> 7;
  int col  = idx & (D - 1);
  float s = dis[node];
  out[idx] = bias[col] + H[idx] * s * s;
}

// one wave per edge; lane handles a float4 (4 cols): out[dst] += norm * H[src]
__global__ __launch_bounds__(256) void k_scatter(const float* __restrict__ H,
                                                 const int* __restrict__ src,
                                                 const int* __restrict__ dst,
                                                 const float* __restrict__ dis,
                                                 float* __restrict__ out,
                                                 int nedges) {
  int e = blockIdx.x * 8 + (threadIdx.x >> 5);
  if (e >= nedges) return;
  int lane = threadIdx.x & 31;
  int s = src[e];
  int d = dst[e];
  float nrm = dis[s] * dis[d];
  float4 v = reinterpret_cast<const float4*>(H + (size_t)s * D)[lane];
  float* o = out + (size_t)d * D + lane * 4;
  atomicAdd(o + 0, v.x * nrm);
  atomicAdd(o + 1, v.y * nrm);
  atomicAdd(o + 2, v.z * nrm);
  atomicAdd(o + 3, v.w * nrm);
}

__global__ __launch_bounds__(256) void k_tanh(float* d, int total) {
  int i = blockIdx.x * 256 + threadIdx.x;
  if (i < total) d[i] = tanhf(d[i]);
}

// ---------------- launch ----------------

extern "C" void kernel_launch(void* const* d_in, const int* in_sizes, int n_in,
                              void* d_out, int out_size, void* d_ws, size_t ws_size,
                              hipStream_t stream) {
  const float* x   = (const float*)d_in[0];
  const int*   ei  = (const int*)d_in[1];   // [2,E] flat: row0 = src, row1 = dst
  const float* W1  = (const float*)d_in[2];
  const float* b1  = (const float*)d_in[3];
  const float* W2  = (const float*)d_in[4];
  const float* b2  = (const float*)d_in[5];

  const int N = in_sizes[0] / D;   // 50000
  const int E = in_sizes[1] / 2;   // 500000
  const int* src = ei;
  const int* dst = ei + E;

  float* ws   = (float*)d_ws;
  size_t npad = ((size_t)N + 127) & ~(size_t)127;
  float* dis  = ws;                       // N (deg -> rsqrt in place)
  float* hbuf = ws + npad;                // N*D  GEMM output
  float* agg  = hbuf + (size_t)N * D;     // N*D  layer-1 activation
  float* out  = (float*)d_out;            // [0:N*D) = h2, [N*D:2N*D) = x

  const int nb_n  = (N + 255) / 256;
  const int nb_e  = (E + 255) / 256;
  const int nb_nd = (N * D + 255) / 256;
  const int nb_gm = (N + 127) / 128;      // 128 rows per block
  const int nb_sc = (E + 7) / 8;          // 8 edges (waves) per block

  // normalization
  k_fill_one<<<nb_n, 256, 0, stream>>>(dis, N);
  k_deg_accum<<<nb_e, 256, 0, stream>>>(dst, dis, E);
  k_rsqrt<<<nb_n, 256, 0, stream>>>(dis, N);

  // layer 1: agg = tanh(b1 + scatter(norm * (x@W1)))
  k_gemm128<<<nb_gm, 256, 0, stream>>>(x, W1, hbuf, N);
  k_agg_init<<<nb_nd, 256, 0, stream>>>(hbuf, dis, b1, agg, N * D);
  k_scatter<<<nb_sc, 256, 0, stream>>>(hbuf, src, dst, dis, agg, E);
  k_tanh<<<nb_nd, 256, 0, stream>>>(agg, N * D);

  // layer 2: out = tanh(b2 + scatter(norm * (h1@W2)))
  k_gemm128<<<nb_gm, 256, 0, stream>>>(agg, W2, hbuf, N);
  k_agg_init<<<nb_nd, 256, 0, stream>>>(hbuf, dis, b2, out, N * D);
  k_scatter<<<nb_sc, 256, 0, stream>>>(hbuf, src, dst, dis, out, E);
  k_tanh<<<nb_nd, 256, 0, stream>>>(out, N * D);

  // second tuple element: identity init embeddings (copy of x)
  hipMemcpyAsync(out + (size_t)N * D, x, (size_t)N * D * sizeof(float),
                 hipMemcpyDeviceToDevice, stream);
}